// FiLMConv_936302871067
// MI455X (gfx1250) — compile-verified
//
#include <hip/hip_runtime.h>

// ---------------------------------------------------------------------------
// FiLMConv, restructured:
//   M'  = x @ W_msg  + b_msg            (50000 x 128)   [WMMA f32 GEMM]
//   F   = x @ W_film + b_film           (50000 x 256)   [WMMA f32 GEMM]
//   out = x @ W_root + b_root           (50000 x 128)   [WMMA f32 GEMM]
//   S[c]   = sum_{e: col(e)=c} M'[row(e)]      (edge scatter, L2 atomics)
//   deg[c] = #edges into c
//   out[c] += gamma(F[c]) * S[c] + deg[c] * beta(F[c])
// ---------------------------------------------------------------------------

typedef __attribute__((ext_vector_type(2))) float v2f;
typedef __attribute__((ext_vector_type(8))) float v8f;

#define N_NODES 50000
#define N_EDGES 800000
#define D_IN    128

// One wave computes one 16x16 output tile. K=128 fully unrolled as 32
// V_WMMA_F32_16X16X4_F32 ops. Block = 256 threads = 8 waves = 8 tiles.
// NCOLS is compile-time so every load in the unrolled loop is
// base-pointer + immediate offset (no per-iteration 64-bit address math).
template <int NCOLS>
__global__ void film_gemm(const float* __restrict__ A,   // Mrows x 128
                          const float* __restrict__ B,   // 128 x NCOLS (row-major)
                          const float* __restrict__ bias,// NCOLS
                          float* __restrict__ C,         // Mrows x NCOLS
                          int Mrows)
{
    const int lane = threadIdx.x & 31;
    const int wave = threadIdx.x >> 5;
    constexpr int ntN = NCOLS >> 4;
    const int tile = blockIdx.x * 8 + wave;
    if (tile >= (Mrows >> 4) * ntN) return;          // wave-uniform: EXEC stays all-1s

    const int m0 = (tile / ntN) << 4;
    const int n0 = (tile % ntN) << 4;

    const int half = lane >> 4;       // 0: K={0,1}, 1: K={2,3} within each k-step
    const int l16  = lane & 15;       // A: M row select, B: N col select
    const int kb   = half << 1;

    // Per-lane base pointers; all loop accesses are immediate offsets off these.
    const float* __restrict__ abase = A + (unsigned)(m0 + l16) * D_IN + kb;
    const float* __restrict__ bbase = B + (unsigned)kb * NCOLS + n0 + l16;

    v8f acc = {};

#pragma unroll
    for (int k0 = 0; k0 < D_IN; k0 += 4) {
        // A fragment: x[m0+l16, k0+kb .. k0+kb+1]  (8B-aligned b64 load, imm offset)
        v2f a = *(const v2f*)(abase + k0);
        // B fragment: W[k0+kb, n0+l16], W[k0+kb+1, n0+l16]  (imm offsets)
        v2f b;
        b.x = bbase[k0 * NCOLS];
        b.y = bbase[k0 * NCOLS + NCOLS];
        acc = __builtin_amdgcn_wmma_f32_16x16x4_f32(
            /*neg_a=*/false, a, /*neg_b=*/false, b,
            /*c_mod=*/(short)0, acc, /*reuse_a=*/false, /*reuse_b=*/false);
    }

    const float bv = bias[n0 + l16];
    float* __restrict__ cbase = C + (unsigned)(m0 + (half << 3)) * NCOLS + n0 + l16;
#pragma unroll
    for (int r = 0; r < 8; ++r) {
        // C/D layout: VGPR r -> M = r + 8*(lane/16), N = lane%16
        cbase[r * NCOLS] = acc[r] + bv;
    }
}

// Zero S (N_NODES*128) and deg (N_NODES) — laid out contiguously in ws.
__global__ void film_zero4(float4* __restrict__ p, int n4)
{
    int i = blockIdx.x * blockDim.x + threadIdx.x;
    if (i < n4) p[i] = make_float4(0.f, 0.f, 0.f, 0.f);
}

// One wave per edge: gather M'[row] (float4/lane, L2-resident) and
// atomically accumulate into S[col]; lane 0 bumps deg[col].
__global__ void film_edge_scatter(const long long* __restrict__ ei, // (2, E)
                                  const float* __restrict__ Mbuf,   // N x 128
                                  float* __restrict__ S,            // N x 128
                                  float* __restrict__ deg,          // N
                                  int nEdges)
{
    const int e    = (blockIdx.x * blockDim.x + threadIdx.x) >> 5;
    const int lane = threadIdx.x & 31;
    if (e >= nEdges) return;

    const long long r = ei[e];
    const long long c = ei[(size_t)nEdges + e];

    const float4 mv = *(const float4*)(Mbuf + (size_t)r * D_IN + lane * 4);
    float* sp = S + (size_t)c * D_IN + lane * 4;
    atomicAdd(sp + 0, mv.x);
    atomicAdd(sp + 1, mv.y);
    atomicAdd(sp + 2, mv.z);
    atomicAdd(sp + 3, mv.w);
    if (lane == 0) atomicAdd(deg + c, 1.0f);
}

// One wave per node: out += gamma * S + deg * beta   (float4/lane)
__global__ void film_finalize(const float* __restrict__ F,   // N x 256 = [gamma|beta] (bias incl.)
                              const float* __restrict__ S,   // N x 128
                              const float* __restrict__ deg, // N
                              float* __restrict__ out,       // N x 128
                              int nNodes)
{
    const int n    = (blockIdx.x * blockDim.x + threadIdx.x) >> 5;
    const int lane = threadIdx.x & 31;
    if (n >= nNodes) return;

    const float  d = deg[n];
    const float4 g = *(const float4*)(F + (size_t)n * 256 +       lane * 4);
    const float4 b = *(const float4*)(F + (size_t)n * 256 + 128 + lane * 4);
    const float4 s = *(const float4*)(S + (size_t)n * D_IN +      lane * 4);

    float4* op = (float4*)(out + (size_t)n * D_IN + lane * 4);
    float4 o = *op;
    o.x += g.x * s.x + d * b.x;
    o.y += g.y * s.y + d * b.y;
    o.z += g.z * s.z + d * b.z;
    o.w += g.w * s.w + d * b.w;
    *op = o;
}

extern "C" void kernel_launch(void* const* d_in, const int* in_sizes, int n_in,
                              void* d_out, int out_size, void* d_ws, size_t ws_size,
                              hipStream_t stream)
{
    const float*     x      = (const float*)d_in[0];
    const long long* ei     = (const long long*)d_in[1];
    const float*     W_msg  = (const float*)d_in[2];
    const float*     b_msg  = (const float*)d_in[3];
    const float*     W_film = (const float*)d_in[4];
    const float*     b_film = (const float*)d_in[5];
    const float*     W_root = (const float*)d_in[6];
    const float*     b_root = (const float*)d_in[7];
    float*           out    = (float*)d_out;

    const int N = N_NODES, E = N_EDGES;

    // Workspace layout (floats): M'[N*128] | F[N*256] | S[N*128] | deg[N]
    float* ws   = (float*)d_ws;
    float* Mbuf = ws;
    float* Fbuf = Mbuf + (size_t)N * 128;
    float* S    = Fbuf + (size_t)N * 256;
    float* deg  = S    + (size_t)N * 128;

    // 1) zero S + deg (contiguous N*129 floats; divisible by 4)
    {
        const int n4 = (N * 129) / 4; // 1,612,500 float4s
        film_zero4<<<(n4 + 255) / 256, 256, 0, stream>>>((float4*)S, n4);
    }

    // 2) node-level GEMMs (f32 WMMA). Tiles: (N/16) * (NCOLS/16); 8 waves/block.
    {
        const int tiles128 = (N / 16) * (128 / 16); // 25000
        const int tiles256 = (N / 16) * (256 / 16); // 50000
        film_gemm<128><<<(tiles128 + 7) / 8, 256, 0, stream>>>(x, W_msg,  b_msg,  Mbuf, N);
        film_gemm<256><<<(tiles256 + 7) / 8, 256, 0, stream>>>(x, W_film, b_film, Fbuf, N);
        film_gemm<128><<<(tiles128 + 7) / 8, 256, 0, stream>>>(x, W_root, b_root, out,  N);
    }

    // 3) edge scatter: one wave per edge (E/8 blocks of 256 threads)
    film_edge_scatter<<<(E + 7) / 8, 256, 0, stream>>>(ei, Mbuf, S, deg, E);

    // 4) finalize per node
    film_finalize<<<(N + 7) / 8, 256, 0, stream>>>(Fbuf, S, deg, out, N);
}